// UpSampling3D_33062658244650
// MI455X (gfx1250) — compile-verified
//
#include <hip/hip_runtime.h>
#include <math.h>

// ---------------------------------------------------------------------------
// Trilinear 2x upsample (align_corners=True), f32.
// in : [2,32,48,48,48]  out : [2,32,96,96,96]
// Store-bandwidth-bound (255 MB traffic, ~1.2 GFLOP @ 23.3 TB/s -> ~11 us):
//   * async global->LDS staging (ASYNCcnt) for the 8-way-reused input
//   * per-thread-invariant interpolation taps hoisted to registers
//   * 128-bit non-temporal streaming stores for the 226 MB output
// ---------------------------------------------------------------------------

typedef float v4f __attribute__((ext_vector_type(4)));
typedef int   v4i __attribute__((__vector_size__(16)));

#define AS1 __attribute__((address_space(1)))
#define AS3 __attribute__((address_space(3)))

#ifndef __has_builtin
#define __has_builtin(x) 0
#endif

#if __has_builtin(__builtin_amdgcn_global_load_async_to_lds_b128)
#define HAVE_ASYNC_LDS 1
#else
#define HAVE_ASYNC_LDS 0
#endif

constexpr int NB_NH     = 2 * 32;         // 64 independent (b,h) volumes
constexpr int NIN       = 48;
constexpr int NOUT      = 96;
constexpr int PLANE     = NIN * NIN;      // 2304 floats per input r-plane
constexpr int RPB       = 8;              // output R rows per block
constexpr int RBLKS     = NOUT / RPB;     // 12
constexpr int SLAB_ROWS = 6;              // max input rows needed for 8 out rows
constexpr int NTHREADS  = 288;            // 9 waves; 288 = 24 d-quads * 12 c-lanes

__global__ __launch_bounds__(NTHREADS)
void upsample3d_lin_48to96(const float* __restrict__ xg, float* __restrict__ outg)
{
    __shared__ __attribute__((aligned(16))) float slab[SLAB_ROWS * PLANE]; // 54 KB
    __shared__ v4f tabv[NOUT]; // {w_lo, w_hi, bitcast(lo_index), 0}

    const int tid  = threadIdx.x;
    const int bh   = blockIdx.x / RBLKS;
    const int rblk = blockIdx.x % RBLKS;
    const int R0   = rblk * RPB;

    const double STEP = 47.0 / 95.0;      // numpy linspace step, single rounding
    const int rbase = (R0 == 0) ? 0 : ((int)ceil((double)R0 * STEP) - 1);

    // ---- stage up to 6 input r-planes into LDS via async B128 copies -------
    const float* src_base = xg + (size_t)bh * (NIN * PLANE);
    constexpr int QR   = PLANE / 4;            // 576 x 16B per plane
    constexpr int QTOT = SLAB_ROWS * QR;       // 3456 (= 12 * 288, no tail)
    for (int i = tid; i < QTOT; i += NTHREADS) {
        const int row = i / QR;
        const int off = i - row * QR;
        int rin = rbase + row;
        if (rin > NIN - 1) rin = NIN - 1;      // clamp (extra rows unused)
        float* g = const_cast<float*>(src_base + (size_t)rin * PLANE) + off * 4;
        float* l = &slab[row * PLANE + off * 4];
#if HAVE_ASYNC_LDS
        __builtin_amdgcn_global_load_async_to_lds_b128(
            (AS1 v4i*)g, (AS3 v4i*)l, /*offset=*/0, /*cpol=*/0);
#else
        *(v4f*)l = *(const v4f*)g;
#endif
    }

    // ---- 48->96 linear-interp table (identical for R, C, D axes) -----------
    // Mirrors numpy float64 weight build incl. +1e-6 normalization & edges.
    if (tid < NOUT) {
        const int i = tid;
        int lo; float a, b;
        if (i == 0) {
            lo = 0;
            a  = (float)(47.0 / (47.0 + 1e-6)); // wrap column gets exact 0 weight
            b  = 0.0f;
        } else {
            const double xx = (i == NOUT - 1) ? 47.0 : (double)i * STEP;
            const int j = (int)ceil(xx);        // == searchsorted(left) here
            lo = j - 1;
            const double dlo  = xx - (double)lo;   // weight on column j
            const double dhi  = (double)j - xx;    // weight on column j-1
            const double norm = (dlo + dhi) + 1e-6;
            a = (float)(dhi / norm);
            b = (float)(dlo / norm);
        }
        v4f e; e.x = a; e.y = b; e.z = __int_as_float(lo); e.w = 0.0f;
        tabv[i] = e;
    }

#if HAVE_ASYNC_LDS
#if __has_builtin(__builtin_amdgcn_s_wait_asynccnt)
    __builtin_amdgcn_s_wait_asynccnt(0);
#else
    asm volatile("s_wait_asynccnt 0" ::: "memory");
#endif
#endif
    __syncthreads();

    // ---- per-thread-invariant D-axis taps ----------------------------------
    const int d4    = tid % 24;      // this thread's D quad, fixed forever
    const int c_off = tid / 24;      // 0..11
    const int Dbase = d4 * 4;

    int   off_d[4];                  // clamped input window offsets
    float Wk[4][4];                  // 4 outputs x 4 window taps
    {
        v4f t0 = tabv[Dbase];
        const int dl0 = __float_as_int(t0.z);
#pragma unroll
        for (int j = 0; j < 4; ++j) {
            int dj = dl0 + j;
            if (dj > NIN - 1) dj = NIN - 1;    // clamp: stray tap has 0 weight
            off_d[j] = dj;
        }
#pragma unroll
        for (int k = 0; k < 4; ++k) {
            v4f t = tabv[Dbase + k];
            const int jk = __float_as_int(t.z) - dl0;   // in [0,2]
#pragma unroll
            for (int j = 0; j < 4; ++j) {
                float w = 0.0f;
                if (j == jk)     w += t.x;
                if (j == jk + 1) w += t.y;
                Wk[k][j] = w;
            }
        }
    }

    // ---- per-thread C entries (8 columns, reused across all 8 R rows) ------
    float ac[8], bc[8];
    int   cbase[8];
#pragma unroll
    for (int cc = 0; cc < 8; ++cc) {
        v4f t = tabv[cc * 12 + c_off];
        ac[cc]    = t.x;
        bc[cc]    = t.y;
        cbase[cc] = __float_as_int(t.z) * NIN;
    }

    // ---- produce 8 x 96 x 96 outputs ---------------------------------------
    float* dst_bh = outg + (size_t)bh * ((size_t)NOUT * NOUT * NOUT);
    for (int Ro = 0; Ro < RPB; ++Ro) {
        const int R = R0 + Ro;
        v4f tr = tabv[R];
        const float ar = tr.x, br = tr.y;
        const int   rl = __float_as_int(tr.z) - rbase;
        const float* blo = &slab[rl * PLANE];
        const float* bhi = blo + PLANE;
        float* drow = dst_bh + (size_t)R * NOUT * NOUT + Dbase;

#pragma unroll
        for (int cc = 0; cc < 8; ++cc) {
            const float* s00 = blo + cbase[cc];
            const float* s01 = s00 + NIN;
            const float* s10 = bhi + cbase[cc];
            const float* s11 = s10 + NIN;

            const float w00 = ar * ac[cc], w01 = ar * bc[cc];
            const float w10 = br * ac[cc], w11 = br * bc[cc];

            float bl[4];                       // bilinear over the d-window
#pragma unroll
            for (int j = 0; j < 4; ++j) {
                const int o = off_d[j];
                bl[j] = w00 * s00[o] + w01 * s01[o] + w10 * s10[o] + w11 * s11[o];
            }

            v4f ov;
#pragma unroll
            for (int k = 0; k < 4; ++k)
                ov[k] = Wk[k][0] * bl[0] + Wk[k][1] * bl[1]
                      + Wk[k][2] * bl[2] + Wk[k][3] * bl[3];

            v4f* dst = (v4f*)(drow + (size_t)(cc * 12 + c_off) * NOUT);
            __builtin_nontemporal_store(ov, dst);   // streaming B128 store
        }
    }
}

extern "C" void kernel_launch(void* const* d_in, const int* in_sizes, int n_in,
                              void* d_out, int out_size, void* d_ws, size_t ws_size,
                              hipStream_t stream)
{
    (void)in_sizes; (void)n_in; (void)out_size; (void)d_ws; (void)ws_size;
    const float* x   = (const float*)d_in[0];
    float*       out = (float*)d_out;
    dim3 grid(NB_NH * RBLKS);   // 768 workgroups
    dim3 block(NTHREADS);       // 9 waves (wave32)
    upsample3d_lin_48to96<<<grid, block, 0, stream>>>(x, out);
}